// Attn_layer_39462159515865
// MI455X (gfx1250) — compile-verified
//
#include <hip/hip_runtime.h>
#include <math.h>

#define N_NODES 8192
#define ODIM    64
#define NEDGE   262144

typedef __attribute__((ext_vector_type(2))) float v2f;
typedef __attribute__((ext_vector_type(8))) float v8f;

// LDS strides padded to 68 floats: keeps float4 stores 16B-aligned (68*4 % 16 == 0)
// and rotates banks by 4 per row so the 16-lane column reads of the A fragment
// hit 16 distinct banks.
#define XS_STRIDE 68
#define WS_STRIDE 68

// ---------------------------------------------------------------------------
// Kernel 1: y = x @ W  via V_WMMA_F32_16X16X4_F32 (exact fp32, bandwidth-bound)
// block = 256 threads = 8 waves; block computes 32 rows x 64 cols of y.
// wave w: row-tile = w/4 (16 rows), col-tile = w%4 (16 cols).
// ---------------------------------------------------------------------------
__global__ __launch_bounds__(256) void gemm_xw_wmma(const float* __restrict__ x,
                                                    const float* __restrict__ w,
                                                    float* __restrict__ y) {
    __shared__ float xs[32 * XS_STRIDE];   // 32 rows of x, 64-wide K chunk
    __shared__ float ws[64 * WS_STRIDE];   // 64 K rows of W, 64 cols

    const int t     = threadIdx.x;
    const int waveI = t >> 5;
    const int lane  = t & 31;
    const int half  = lane >> 4;   // 0: K pair {0,1}; 1: K pair {2,3} (A and B layouts)
    const int mrow  = lane & 15;   // M index for A, N index for B/C
    const int n0    = (waveI & 3) * 16;
    const int rtile = waveI >> 2;  // 0..1
    const int r0    = blockIdx.x * 32;

    v8f c = {};

    for (int kk = 0; kk < N_NODES; kk += 64) {
        __syncthreads();
        // ---- stage x chunk: rows [r0, r0+32), cols [kk, kk+64) : 512 float4
#pragma unroll
        for (int j = 0; j < 2; ++j) {
            int f   = t + j * 256;          // float4 index 0..511
            int row = f >> 4;               // 0..31
            int c4  = (f & 15) << 2;        // 0,4,...,60
            const float4 v = *(const float4*)(x + (size_t)(r0 + row) * N_NODES + kk + c4);
            *(float4*)&xs[row * XS_STRIDE + c4] = v;
            if (kk + 64 < N_NODES)          // prefetch next chunk of the x stream
                __builtin_prefetch(x + (size_t)(r0 + row) * N_NODES + kk + 64 + c4, 0, 1);
        }
        // ---- stage W chunk: rows [kk, kk+64), cols [0,64) : 1024 float4
#pragma unroll
        for (int j = 0; j < 4; ++j) {
            int f   = t + j * 256;          // 0..1023
            int row = f >> 4;               // 0..63
            int c4  = (f & 15) << 2;
            const float4 v = *(const float4*)(w + (size_t)(kk + row) * ODIM + c4);
            *(float4*)&ws[row * WS_STRIDE + c4] = v;
        }
        __syncthreads();

        const float* xr = &xs[(rtile * 16 + mrow) * XS_STRIDE];
#pragma unroll
        for (int kb = 0; kb < 64; kb += 4) {
            // A 16x4 f32 layout: lane<16 -> K={kb,kb+1}; lane>=16 -> K={kb+2,kb+3}
            const int k0 = kb + half * 2;
            v2f a;
            a[0] = xr[k0];
            a[1] = xr[k0 + 1];
            // B 4x16 f32 layout mirrors A: VGPR0 = rows K0|K2, VGPR1 = rows K1|K3
            v2f b;
            b[0] = ws[(size_t)k0 * WS_STRIDE + n0 + mrow];
            b[1] = ws[(size_t)(k0 + 1) * WS_STRIDE + n0 + mrow];
            c = __builtin_amdgcn_wmma_f32_16x16x4_f32(
                    /*neg_a=*/false, a, /*neg_b=*/false, b,
                    /*c_mod=*/(short)0, c, /*reuse_a=*/false, /*reuse_b=*/false);
        }
    }

    // C/D 16x16 f32 layout: VGPR j -> M = j (lanes 0-15) / j+8 (lanes 16-31), N = mrow
    const int rbase = r0 + rtile * 16 + half * 8;
#pragma unroll
    for (int j = 0; j < 8; ++j)
        y[(size_t)(rbase + j) * ODIM + n0 + mrow] = c[j];
}

// ---------------------------------------------------------------------------
// Kernel 2: sum of squares of attn (grid-stride float4 stream, 256 MB)
// ---------------------------------------------------------------------------
__global__ __launch_bounds__(256) void norm_sq_kernel(const float* __restrict__ a,
                                                      float* __restrict__ accum,
                                                      long long n4) {
    long long i      = (long long)blockIdx.x * blockDim.x + threadIdx.x;
    long long stride = (long long)gridDim.x * blockDim.x;
    const float4* a4 = (const float4*)a;
    float s = 0.0f;
    for (; i < n4; i += stride) {
        float4 v = a4[i];
        s += v.x * v.x + v.y * v.y + v.z * v.z + v.w * v.w;
    }
    // wave32 reduction
    for (int off = 16; off > 0; off >>= 1)
        s += __shfl_down(s, off, 32);
    __shared__ float red[8];
    const int lane = threadIdx.x & 31;
    const int wv   = threadIdx.x >> 5;
    if (lane == 0) red[wv] = s;
    __syncthreads();
    if (threadIdx.x == 0) {
        float tot = 0.0f;
#pragma unroll
        for (int k = 0; k < 8; ++k) tot += red[k];
        atomicAdd(accum, tot);
    }
}

// ---------------------------------------------------------------------------
// Kernel 3: edge scatter  out[row[e]] += values[e]*attn[row[e],col[e]] * y[col[e]]
// One wave per edge; lane L handles output columns L and L+32.
// ---------------------------------------------------------------------------
__global__ __launch_bounds__(256) void edge_spmm_kernel(const float* __restrict__ values,
                                                        const int*   __restrict__ row,
                                                        const int*   __restrict__ col,
                                                        const float* __restrict__ attn,
                                                        const float* __restrict__ y,
                                                        float* __restrict__ out) {
    const int e    = (int)((blockIdx.x * blockDim.x + threadIdx.x) >> 5);
    const int lane = threadIdx.x & 31;
    if (e >= NEDGE) return;
    const int r  = row[e];
    const int cc = col[e];
    // Uniform-per-wave scalar loads (hardware/compiler collapse to one request).
    const float ev = values[e] * attn[(size_t)r * N_NODES + cc];
    const float* yr = y + (size_t)cc * ODIM;
    float* orow     = out + (size_t)r * ODIM;
    atomicAdd(&orow[lane],      ev * yr[lane]);
    atomicAdd(&orow[lane + 32], ev * yr[lane + 32]);
}

// ---------------------------------------------------------------------------
// Kernel 4: write sqrt(sum-of-squares) into the scalar output slot
// ---------------------------------------------------------------------------
__global__ void finalize_norm_kernel(const float* __restrict__ accum,
                                     float* __restrict__ out_scalar) {
    if (threadIdx.x == 0 && blockIdx.x == 0)
        *out_scalar = sqrtf(*accum);
}

// ---------------------------------------------------------------------------
extern "C" void kernel_launch(void* const* d_in, const int* in_sizes, int n_in,
                              void* d_out, int out_size, void* d_ws, size_t ws_size,
                              hipStream_t stream) {
    const float* x      = (const float*)d_in[0];   // [8192, 8192]
    const float* attn   = (const float*)d_in[1];   // [8192, 8192]
    const float* weight = (const float*)d_in[2];   // [8192, 64]
    const float* values = (const float*)d_in[3];   // [E]
    const int*   row    = (const int*)d_in[4];     // [E]
    const int*   col    = (const int*)d_in[5];     // [E]
    float* out = (float*)d_out;                    // [8192*64] ++ [1]

    float* y     = (float*)d_ws;                                          // 2 MB scratch
    float* accum = (float*)((char*)d_ws + (size_t)N_NODES * ODIM * sizeof(float));

    // Atomically-accumulated buffers must be zeroed every call (graph-replay safe).
    hipMemsetAsync(d_out, 0, (size_t)out_size * sizeof(float), stream);
    hipMemsetAsync(accum, 0, sizeof(float), stream);

    gemm_xw_wmma<<<N_NODES / 32, 256, 0, stream>>>(x, weight, y);
    norm_sq_kernel<<<1024, 256, 0, stream>>>(attn, accum,
                                             (long long)N_NODES * N_NODES / 4);
    edge_spmm_kernel<<<(NEDGE * 32) / 256, 256, 0, stream>>>(values, row, col,
                                                             attn, y, out);
    finalize_norm_kernel<<<1, 32, 0, stream>>>(accum, out + (size_t)N_NODES * ODIM);
}